// ContrastiveMoCoKnnInitByBert_53369263620110
// MI455X (gfx1250) — compile-verified
//
#include <hip/hip_runtime.h>
#include <math.h>

// ---------------------------------------------------------------------------
// ContrastiveMoCoKnn:  sim = Q @ Fq^T  -> masked top-k (pos / neg) -> logits
// B=128, K=65536, C=768, top_k=25, T=0.3
// ---------------------------------------------------------------------------

typedef __attribute__((ext_vector_type(2))) float v2f;
typedef __attribute__((ext_vector_type(8))) float v8f;
typedef __attribute__((ext_vector_type(4))) unsigned int u32x4;
typedef __attribute__((ext_vector_type(8))) int i32x8;
typedef __attribute__((ext_vector_type(4))) int i32x4;

#define B_DIM   128
#define K_DIM   65536
#define C_DIM   768
#define TOPK    25
#define REPS    (TOPK + 1)          // pos_min >> 25 for K=64Ki, L=64
#define KT      8                   // K-tiles (16 cols) per workgroup
// TDM pad: +1 DWORD every 256 DWORDs -> effective LDS row stride 771 floats
// (771 mod 64 = 3 -> 16 column-reading lanes hit 16 distinct banks)
#define ROW_STRIDE 771
#define TILE_F32   (16 * ROW_STRIDE)            // floats per LDS buffer
#define PAD_IDX(c) ((c) + ((c) >> 8))           // TDM padding address map

// ---------------------------------------------------------------------------
// TDM: DMA one 16x768 f32 feature tile (row-major, stride 768) into LDS with
// padding, per CDNA5 D# groups 0/1 (CDNA5 ISA).
// 6-arg builtin (clang-23 / therock): (g0, g1, g2, g3, g4, cpol).
// ---------------------------------------------------------------------------
__device__ __forceinline__ void tdm_load_tile(const float* gsrc, unsigned lds_byte_off)
{
#if __has_builtin(__builtin_amdgcn_tensor_load_to_lds)
    unsigned long long ga = (unsigned long long)(uintptr_t)gsrc;
    u32x4 g0;
    g0.x = 0x1u;                                   // count=1, user descriptor
    g0.y = lds_byte_off;                           // lds_addr (bytes)
    g0.z = (unsigned)(ga & 0xFFFFFFFFu);           // global_addr[31:0]
    g0.w = (unsigned)((ga >> 32) & 0x01FFFFFFu)    // global_addr[56:32]
         | (2u << 30);                             // type = 2 ("image")

    i32x8 g1;
    g1[0] = (2 << 16)                              // data_size = 4B
          | (1 << 20)                              // pad_enable
          | (7 << 22);                             // pad_interval = 256 DW
                                                   // pad_amount  = 1 DW (enc 0)
    g1[1] = (int)((C_DIM & 0xFFFFu) << 16);        // tensor_dim0.lo16 << 16
    g1[2] = (int)((C_DIM >> 16) |                  // tensor_dim0.hi16
                  ((K_DIM & 0xFFFFu) << 16));      // tensor_dim1.lo16
    g1[3] = (int)((K_DIM >> 16) |                  // tensor_dim1.hi16
                  (C_DIM << 16));                  // tile_dim0 = 768
    g1[4] = 16;                                    // tile_dim1 = 16, tile_dim2 = 0
    g1[5] = C_DIM;                                 // tensor_dim0_stride.lo32 = 768
    g1[6] = 0;                                     // stride.hi16 | dim1_stride.lo16
    g1[7] = 0;
    i32x4 z4 = {};
    i32x8 z8 = {};
    __builtin_amdgcn_tensor_load_to_lds(g0, g1, z4, z4, z8, /*cpol=*/0);
#else
    (void)gsrc; (void)lds_byte_off;
#endif
}

__device__ __forceinline__ void tdm_wait(int outstanding)
{
#if __has_builtin(__builtin_amdgcn_s_wait_tensorcnt)
    if (outstanding) __builtin_amdgcn_s_wait_tensorcnt(1);
    else             __builtin_amdgcn_s_wait_tensorcnt(0);
#else
    __builtin_amdgcn_s_barrier();
#endif
}

// ---------------------------------------------------------------------------
// Kernel 1: sim[128,65536] = Q[128,768] @ F[65536,768]^T using f32 WMMA.
// 1 workgroup (8 waves) handles KT=8 consecutive 16-col tiles; feature tiles
// are TDM-DMA'd into double-buffered LDS and reused by all 8 waves.
// ---------------------------------------------------------------------------
__global__ __launch_bounds__(256) void gemm_sim_kernel(
    const float* __restrict__ q,    // [128, 768]
    const float* __restrict__ f,    // [65536, 768]
    float* __restrict__ sim)        // [128, 65536]
{
    __shared__ float bs[2 * TILE_F32];              // 2 x 49,344 B tile buffers

    const int tid   = threadIdx.x;
    const int wave  = tid >> 5;                     // wave32: 8 waves / block
    const int lane  = tid & 31;
    const int m0    = wave * 16;                    // 16 rows of Q per wave
    const int nrow  = lane & 15;                    // N (and M) sub-index
    const int half  = lane >> 4;                    // lane half selector
    const int kwg   = blockIdx.x * (16 * KT);       // first column of this WG

    const unsigned lds_base = (unsigned)(uintptr_t)&bs[0];
    const float* arow = q + (size_t)(m0 + nrow) * C_DIM;

#if __has_builtin(__builtin_amdgcn_tensor_load_to_lds)
    const bool use_tdm = true;
#else
    const bool use_tdm = false;
#endif

    // prologue: DMA tile 0 into buffer 0
    if (use_tdm) {
        if (tid < 32)
            tdm_load_tile(f + (size_t)kwg * C_DIM, lds_base);
    }

    for (int t = 0; t < KT; ++t) {
        const int kbase = kwg + t * 16;
        float* buf = &bs[(t & 1) * TILE_F32];

        if (use_tdm) {
            if (tid < 32) {
                if (t + 1 < KT) {
                    tdm_load_tile(f + (size_t)(kbase + 16) * C_DIM,
                                  lds_base + ((t + 1) & 1) * TILE_F32 * 4u);
                    tdm_wait(1);            // tile t landed; t+1 in flight
                } else {
                    tdm_wait(0);            // last tile landed
                }
            }
        } else {
            // fallback: synchronous staged copy (same padded layout)
            #pragma unroll
            for (int it = 0; it < 12; ++it) {
                int c    = tid + 256 * it;          // chunk id 0..3071
                int row  = c / 192;                 // 192 float4 per row
                int col  = (c % 192) * 4;
                float4 v = *(const float4*)(f + (size_t)(kbase + row) * C_DIM + col);
                *(float4*)(&buf[row * ROW_STRIDE + PAD_IDX(col)]) = v;
            }
        }
        __syncthreads();                            // tile visible to all waves

        v8f acc = {};
        const float* brow = &buf[nrow * ROW_STRIDE];
        #pragma unroll 4
        for (int c0 = 0; c0 < C_DIM; c0 += 4) {
            // A fragment: 16x4 f32. VGPR0: K=c0+2h, VGPR1: K=c0+2h+1.
            v2f a, b;
            a.x = arow[c0 + 2 * half + 0];
            a.y = arow[c0 + 2 * half + 1];
            // B fragment: 4x16 f32 (B[c][n] = F[kbase+n][c]) from padded LDS.
            int c1 = c0 + half, c2 = c0 + 2 + half;
            b.x = brow[PAD_IDX(c1)];
            b.y = brow[PAD_IDX(c2)];

            __builtin_prefetch(arow + c0 + 64, 0, 3);   // global_prefetch_b8

            acc = __builtin_amdgcn_wmma_f32_16x16x4_f32(
                      /*neg_a=*/false, a, /*neg_b=*/false, b,
                      /*c_mod=*/(short)0, acc,
                      /*reuse_a=*/false, /*reuse_b=*/false);
        }

        // C/D layout: VGPR v: lanes0-15 -> M=v, lanes16-31 -> M=v+8; N=lane&15
        #pragma unroll
        for (int v = 0; v < 8; ++v) {
            int m = m0 + v + 8 * half;
            sim[(size_t)m * K_DIM + kbase + nrow] = acc[v];
        }
        __syncthreads();                            // buffer free for next DMA
    }
}

// ---------------------------------------------------------------------------
// Kernel 2: per-row positive counts.
// ---------------------------------------------------------------------------
__global__ __launch_bounds__(256) void count_pos_kernel(
    const int* __restrict__ label_q,      // [128]
    const int* __restrict__ label_queue,  // [65536]
    int* __restrict__ cnt)                // [128] out
{
    __shared__ int s[256];
    const int b  = blockIdx.x;
    const int my = label_q[b];
    int c = 0;
    for (int k = threadIdx.x; k < K_DIM; k += 256)
        c += (label_queue[k] == my) ? 1 : 0;
    s[threadIdx.x] = c;
    __syncthreads();
    for (int off = 128; off; off >>= 1) {
        if (threadIdx.x < off) s[threadIdx.x] += s[threadIdx.x + off];
        __syncthreads();
    }
    if (threadIdx.x == 0) cnt[b] = s[0];
}

// Kernel 3: pos_min = min over rows -> cnt[128]
__global__ __launch_bounds__(128) void min_pos_kernel(int* __restrict__ cnt)
{
    __shared__ int s[128];
    s[threadIdx.x] = cnt[threadIdx.x];
    __syncthreads();
    for (int off = 64; off; off >>= 1) {
        if (threadIdx.x < off)
            s[threadIdx.x] = min(s[threadIdx.x], s[threadIdx.x + off]);
        __syncthreads();
    }
    if (threadIdx.x == 0) cnt[B_DIM] = s[0];
}

// ---------------------------------------------------------------------------
// In-LDS bitonic sort, descending, N = 65536 f32 (256 KB dynamic LDS).
// ---------------------------------------------------------------------------
__device__ __forceinline__ void bitonic_sort_desc(float* s)
{
    const int N = K_DIM;
    for (int k = 2; k <= N; k <<= 1) {
        for (int j = k >> 1; j > 0; j >>= 1) {
            for (int i = threadIdx.x; i < N; i += 1024) {
                int ixj = i ^ j;
                if (ixj > i) {
                    float a = s[i], c = s[ixj];
                    bool up = ((i & k) == 0);
                    // descending overall
                    if ((up && a < c) || (!up && a > c)) {
                        s[i] = c;
                        s[ixj] = a;
                    }
                }
            }
            __syncthreads();
        }
    }
}

// ---------------------------------------------------------------------------
// Kernel 4: per-row (one WG per row): sort positives -> pos_cat[26],
// sort negatives -> stream out 26 x (1 + neg_min) logits / T.
// ---------------------------------------------------------------------------
__global__ __launch_bounds__(1024) void sort_write_kernel(
    const float* __restrict__ sim,        // [128, 65536]
    const int* __restrict__ label_q,
    const int* __restrict__ label_queue,
    const int* __restrict__ cnt,          // cnt[128] = pos_min
    float* __restrict__ out,
    long long neg_min)
{
    extern __shared__ float s[];          // 65536 f32
    __shared__ float pos_cat[REPS];

    const int b  = blockIdx.x;
    const int my = label_q[b];
    const float* row = sim + (size_t)b * K_DIM;

    // ---- pass 1: positives (mask negatives to -inf), sort desc ----------
    for (int k = threadIdx.x; k < K_DIM; k += 1024)
        s[k] = (label_queue[k] == my) ? row[k] : -INFINITY;
    __syncthreads();
    bitonic_sort_desc(s);

    const int pos_min = cnt[B_DIM];
    const int pc = (pos_min < TOPK) ? pos_min : TOPK;
    if (threadIdx.x < REPS) {
        int j = threadIdx.x;
        pos_cat[j] = (j < pc) ? s[j] : s[pos_min - 1];
    }
    __syncthreads();

    // ---- pass 2: negatives (mask positives to -inf), sort desc ----------
    for (int k = threadIdx.x; k < K_DIM; k += 1024)
        s[k] = (label_queue[k] == my) ? -INFINITY : row[k];
    __syncthreads();
    bitonic_sort_desc(s);

    // ---- write logits: 26 rows of [pos_cat[j], neg_top...] / T ----------
    const float invT = 1.0f / 0.3f;
    const size_t rowlen = (size_t)neg_min + 1;
    for (int j = 0; j < REPS; ++j) {
        float* orow = out + ((size_t)b * REPS + j) * rowlen;
        if (threadIdx.x == 0) orow[0] = pos_cat[j] * invT;
        for (long long i = threadIdx.x; i < neg_min; i += 1024)
            orow[1 + i] = s[i] * invT;
    }
}

// ---------------------------------------------------------------------------
// Launch
// ---------------------------------------------------------------------------
extern "C" void kernel_launch(void* const* d_in, const int* in_sizes, int n_in,
                              void* d_out, int out_size, void* d_ws, size_t ws_size,
                              hipStream_t stream)
{
    const float* q  = (const float*)d_in[0];   // liner_q      [128,768]
    const float* f  = (const float*)d_in[1];   // feature_queue[65536,768]
    const int*   lq = (const int*)d_in[2];     // label_q      [128]
    const int*   lQ = (const int*)d_in[3];     // label_queue  [65536]
    // d_in[4] = top_k (25, hardcoded as TOPK)

    float* sim = (float*)d_ws;                                    // 32 MB
    int*   cnt = (int*)((char*)d_ws + (size_t)B_DIM * K_DIM * 4); // 129 ints

    // neg_min recovered from the (data-dependent) output geometry.
    long long neg_min = (long long)out_size / ((long long)B_DIM * REPS) - 1;

    // 1) GEMM: 512 WGs x 8 double-buffered TDM-staged K-tiles.
    gemm_sim_kernel<<<K_DIM / (16 * KT), 256, 0, stream>>>(q, f, sim);

    // 2) positive counts + pos_min
    count_pos_kernel<<<B_DIM, 256, 0, stream>>>(lq, lQ, cnt);
    min_pos_kernel<<<1, 128, 0, stream>>>(cnt);

    // 3) per-row masked sorts + output streaming (256 KB dynamic LDS)
    (void)hipFuncSetAttribute((const void*)sort_write_kernel,
                              hipFuncAttributeMaxDynamicSharedMemorySize,
                              K_DIM * (int)sizeof(float));
    sort_write_kernel<<<B_DIM, 1024, K_DIM * sizeof(float), stream>>>(
        sim, lq, lQ, cnt, (float*)d_out, neg_min);
}